// BaseFisheyeLSSTransform_60129542943
// MI455X (gfx1250) — compile-verified
//
#include <hip/hip_runtime.h>
#include <cstddef>
#include <cstdint>

// ---- problem constants (from reference) ----
#define Bc   2
#define Nc   4
#define Cc   80
#define Dc   59
#define FHc  40
#define FWc  60
#define NXc  360
#define NYc  360
#define NZc  1
#define NUMc   (Bc * NZc * NXc * NYc)          // 259200 voxels
#define NPTSc  (Bc * Nc * Dc * FHc * FWc)      // 1132800 frustum points
#define OUTNc  (Bc * NZc * Cc * NXc * NYc)     // 20736000 output f32
#define YT   40                                 // y-tile in finalize (360 = 9*40)

// TDM pad: +1 DWORD after every 16 DWORDs -> padded(t) = t + (t>>4); row pitch 85
#define TILE_PAD(t) ((t) + ((t) >> 4))
#define TILE_WORDS  (TILE_PAD(YT * Cc - 1) + 1)   // 3399

#if defined(__HIP_DEVICE_COMPILE__) && \
    __has_builtin(__builtin_amdgcn_tensor_load_to_lds) && \
    __has_builtin(__builtin_amdgcn_s_wait_tensorcnt)
#define USE_TDM 1
#else
#define USE_TDM 0
#endif

typedef unsigned int u32;
typedef unsigned int v4u __attribute__((ext_vector_type(4)));
typedef int          v8i __attribute__((ext_vector_type(8)));
typedef int          v4i __attribute__((ext_vector_type(4)));

__device__ __forceinline__ void atomAddF(float* p, float v) {
#if defined(__HIP_DEVICE_COMPILE__)
  unsafeAtomicAdd(p, v);   // hardware global_atomic_add_f32 (no-return form)
#else
  atomicAdd(p, v);
#endif
}

#if USE_TDM
__device__ __forceinline__ u32 ldsOffsetOf(const void* p) {
  // generic -> LDS addrspace cast yields the 32-bit LDS byte offset
  return (u32)(uintptr_t)(const __attribute__((address_space(3))) void*)p;
}
#endif

// Vectorized clear: n must be a multiple of 4 (all our sizes are).
__global__ __launch_bounds__(256) void zero_f32_v4(float4* __restrict__ p, int n4) {
  int i = blockIdx.x * blockDim.x + threadIdx.x;
  if (i < n4) p[i] = make_float4(0.f, 0.f, 0.f, 0.f);
}

// Compute voxel id for a frustum point; returns kept flag.
__device__ __forceinline__ bool geom_voxel(int point,
    const float* __restrict__ rots, const float* __restrict__ trans,
    int& b_out, int& voxel_out)
{
  int w = point % FWc;
  int t = point / FWc;
  int h = t % FHc;  t /= FHc;
  int d = t % Dc;   t /= Dc;
  int n = t % Nc;
  int b = t / Nc;
  b_out = b;

  // frustum: ds in [1..59], az linspace(-1.92,1.92,60), el linspace(-0.61,0.61,40)
  float ds = 1.0f + (float)d;
  float az = -1.92f + (float)w * (3.84f / 59.0f);
  float el = -0.61f + (float)h * (1.22f / 39.0f);
  float ce = cosf(el), se = sinf(el);
  float sa = sinf(az), ca = cosf(az);
  float fx = ds * ce * sa;
  float fy = ds * se;
  float fz = ds * ce * ca;

  const float* R = rots  + (size_t)(b * Nc + n) * 9;
  const float* T = trans + (size_t)(b * Nc + n) * 3;
  float gx = R[0]*fx + R[1]*fy + R[2]*fz + T[0];
  float gy = R[3]*fx + R[4]*fy + R[5]*fz + T[1];
  float gz = R[6]*fx + R[7]*fy + R[8]*fz + T[2];

  // (geom - origin)/dx, truncation toward zero == astype(int32)
  int cx = (int)((gx + 54.0f) / 0.3f);
  int cy = (int)((gy + 54.0f) / 0.3f);
  int cz = (int)((gz + 5.0f)  / 8.0f);

  bool kept = (cx >= 0) & (cx < NXc) & (cy >= 0) & (cy < NYc) & (cz == 0);
  voxel_out = (b * NXc + cx) * NYc + cy;   // NZ==1, cz==0 when kept
  return kept;
}

// One wave32 per point; lanes cover channels (coalesced loads + coalesced atomics).
__global__ __launch_bounds__(256) void scatter_sum(
    const float* __restrict__ xf,
    const float* __restrict__ rots,
    const float* __restrict__ trans,
    float* __restrict__ sums,   // [NUMc][Cc]
    float* __restrict__ cnt)    // [NUMc]
{
  int wave = blockIdx.x * (blockDim.x >> 5) + (threadIdx.x >> 5);
  int lane = threadIdx.x & 31;
  if (wave >= NPTSc) return;

  int b, voxel;
  bool kept = geom_voxel(wave, rots, trans, b, voxel);
  if (!kept) return;

  size_t base = (size_t)wave * Cc;
  // CDNA5 global_prefetch_b8: pull the 320B feature row while atomics queue up
  __builtin_prefetch((const void*)(xf + base + (size_t)lane * 10), 0, 0);

  if (lane == 0) atomAddF(&cnt[voxel], 1.0f);
  size_t sb = (size_t)voxel * Cc;
  atomAddF(&sums[sb + lane],      xf[base + lane]);
  atomAddF(&sums[sb + lane + 32], xf[base + lane + 32]);
  if (lane < 16)
    atomAddF(&sums[sb + lane + 64], xf[base + lane + 64]);
}

// Mean + (voxel,c) -> (b,c,x,y) transpose via padded LDS tile.
// Tile load uses the Tensor Data Mover (async DMA, TENSORcnt) when available.
__global__ __launch_bounds__(256) void finalize_t(
    const float* __restrict__ sums,
    const float* __restrict__ cnt,
    float* __restrict__ out)
{
  __shared__ float tile[TILE_WORDS];      // padded: pitch 85 -> bank-conflict-free
  __shared__ float cmax[YT];

  int blk = blockIdx.x;
  int yt = blk % (NYc / YT);
  int x  = (blk / (NYc / YT)) % NXc;
  int b  = blk / ((NYc / YT) * NXc);
  int y0 = yt * YT;

  size_t vox0 = ((size_t)(b * NXc + x)) * NYc + y0;
  const float* src = sums + vox0 * Cc;    // YT*Cc contiguous floats

#if USE_TDM
  if ((threadIdx.x >> 5) == 0) {          // one wave issues the tensor DMA
    uint64_t gaddr = (uint64_t)(uintptr_t)src;
    u32 laddr = ldsOffsetOf(tile);
    // ---- D# group 0 (ISA 08 §8.3): count=1 | lds_addr | global_addr | type=2
    v4u g0;
    g0.x = 1u;                                       // count=1, user mode
    g0.y = laddr;                                    // lds_addr [63:32]
    g0.z = (u32)(gaddr & 0xFFFFFFFFu);               // global_addr[31:0]
    g0.w = (u32)((gaddr >> 32) & 0x01FFFFFFu) | (2u << 30);  // ga[56:32] | type=2
    // ---- D# group 1 (§8.4): 2D tile 80x40 f32, pad 1 DW per 16 DW
    v8i g1;
    g1[0] = (int)((2u << 16)        // data_size = 4 bytes
                | (1u << 20)        // pad_enable
                | (3u << 22));      // pad_interval: 16 DWORDs (pad_amount=0 -> 1 DW)
    g1[1] = (int)((u32)Cc << 16);   // tensor_dim0[15:0]=80 (barrier addr 0)
    g1[2] = (int)((u32)YT << 16);   // tensor_dim0[31:16]=0 | tensor_dim1[15:0]=40
    g1[3] = (int)((u32)Cc << 16);   // tensor_dim1[31:16]=0 | tile_dim0=80
    g1[4] = (int)(u32)YT;           // tile_dim1=40 | tile_dim2=0
    g1[5] = (int)(u32)Cc;           // tensor_dim0_stride[31:0]=80 (rows contiguous)
    g1[6] = 0;                      // stride0[47:32]=0 | stride1[15:0]=0
    g1[7] = 0;                      // stride1[47:16]=0
    v4i gz4 = {0, 0, 0, 0};         // groups 2/3 unused (2D tile)
    v8i gz8 = {0, 0, 0, 0, 0, 0, 0, 0};
    // 6-arg form (clang-23 / therock-10 toolchain)
    __builtin_amdgcn_tensor_load_to_lds(g0, g1, gz4, gz4, gz8, 0);
  }
#else
  for (int t = threadIdx.x; t < YT * Cc; t += blockDim.x)
    tile[TILE_PAD(t)] = src[t];
#endif
  // count loads overlap the in-flight tensor DMA
  for (int t = threadIdx.x; t < YT; t += blockDim.x)
    cmax[t] = fmaxf(cnt[vox0 + t], 1.0f);
#if USE_TDM
  if ((threadIdx.x >> 5) == 0)
    __builtin_amdgcn_s_wait_tensorcnt(0);  // only the issuing wave has TENSORcnt>0
#endif
  __syncthreads();

  size_t dstBase = (size_t)b * Cc * NXc * NYc + (size_t)x * NYc + y0;
  for (int t = threadIdx.x; t < YT * Cc; t += blockDim.x) {
    int c = t / YT, y = t % YT;             // y fastest -> coalesced stores
    int lin = y * Cc + c;                   // padded pitch 85 -> no bank conflicts
    out[dstBase + (size_t)c * (NXc * NYc) + y] = tile[TILE_PAD(lin)] / cmax[y];
  }
}

// ---- fallback path if d_ws can't hold the 84MB accumulator ----
__global__ __launch_bounds__(256) void scatter_direct(
    const float* __restrict__ xf,
    const float* __restrict__ rots,
    const float* __restrict__ trans,
    float* __restrict__ out,    // (b,c,x,y) layout, pre-zeroed
    float* __restrict__ cnt)
{
  int wave = blockIdx.x * (blockDim.x >> 5) + (threadIdx.x >> 5);
  int lane = threadIdx.x & 31;
  if (wave >= NPTSc) return;
  int b, voxel;
  if (!geom_voxel(wave, rots, trans, b, voxel)) return;
  int xy = voxel - b * (NXc * NYc);       // cx*NY + cy
  size_t ob = (size_t)b * Cc * NXc * NYc + xy;
  size_t base = (size_t)wave * Cc;
  if (lane == 0) atomAddF(&cnt[voxel], 1.0f);
  atomAddF(&out[ob + (size_t)lane * (NXc * NYc)],        xf[base + lane]);
  atomAddF(&out[ob + (size_t)(lane + 32) * (NXc * NYc)], xf[base + lane + 32]);
  if (lane < 16)
    atomAddF(&out[ob + (size_t)(lane + 64) * (NXc * NYc)], xf[base + lane + 64]);
}

__global__ __launch_bounds__(256) void normalize_inplace(
    float* __restrict__ out, const float* __restrict__ cnt)
{
  int i = blockIdx.x * blockDim.x + threadIdx.x;
  if (i >= OUTNc) return;
  int y = i % NYc;
  int t = i / NYc;
  int x = t % NXc;  t /= NXc;
  int b = (t / Cc);
  int voxel = (b * NXc + x) * NYc + y;
  out[i] /= fmaxf(cnt[voxel], 1.0f);
}

extern "C" void kernel_launch(void* const* d_in, const int* in_sizes, int n_in,
                              void* d_out, int out_size, void* d_ws, size_t ws_size,
                              hipStream_t stream) {
  (void)in_sizes; (void)n_in; (void)out_size;
  const float* xf    = (const float*)d_in[0];  // (B,N,D,FH,FW,C) f32
  const float* rots  = (const float*)d_in[1];  // (B,N,3,3) f32
  const float* trans = (const float*)d_in[2];  // (B,N,3) f32
  float* out = (float*)d_out;

  const int scatterBlocks = (NPTSc + 7) / 8;   // 8 waves (256 thr) per block
  const size_t needed = (size_t)NUMc * (Cc + 1) * sizeof(float);

  if (ws_size >= needed) {
    float* sums = (float*)d_ws;
    float* cnt  = sums + (size_t)NUMc * Cc;
    const int n4 = (NUMc * (Cc + 1)) / 4;      // sums + cnt contiguous, /4 exact
    zero_f32_v4<<<(n4 + 255) / 256, 256, 0, stream>>>((float4*)sums, n4);
    scatter_sum<<<scatterBlocks, 256, 0, stream>>>(xf, rots, trans, sums, cnt);
    finalize_t<<<Bc * NXc * (NYc / YT), 256, 0, stream>>>(sums, cnt, out);
  } else {
    float* cnt = (float*)d_ws;                 // needs only ~1MB
    zero_f32_v4<<<(NUMc / 4 + 255) / 256, 256, 0, stream>>>((float4*)cnt, NUMc / 4);
    zero_f32_v4<<<(OUTNc / 4 + 255) / 256, 256, 0, stream>>>((float4*)out, OUTNc / 4);
    scatter_direct<<<scatterBlocks, 256, 0, stream>>>(xf, rots, trans, out, cnt);
    normalize_inplace<<<(OUTNc + 255) / 256, 256, 0, stream>>>(out, cnt);
  }
}